// DiffPoolLike_86371792323182
// MI455X (gfx1250) — compile-verified
//
#include <hip/hip_runtime.h>
#include <hip/hip_bf16.h>

typedef float v2f __attribute__((ext_vector_type(2)));
typedef float v8f __attribute__((ext_vector_type(8)));

// ---------------------------------------------------------------------------
// WMMA f32 GEMM: C[Nrows x NCOLS] = A[Nrows x KDIM] @ B[KDIM x NCOLS]
// One wave per 16x16 output tile; K-loop in steps of 4 using
// V_WMMA_F32_16X16X4_F32 (full f32 precision).
//
// Fragment layouts (CDNA5 ISA 7.12.2, wave32):
//   A 16x4 : lane m=L&15, half=L>>4; VGPR0=K(2*half), VGPR1=K(2*half+1)
//   B 4x16 : lane n=L&15, half=L>>4; VGPR0=K(2*half) row, VGPR1=K(2*half+1)
//   C 16x16: VGPR v -> row (v + 8*half), col = L&15
// ---------------------------------------------------------------------------
template <int KDIM, int NCOLS>
__global__ void k_gemm_wmma(const float* __restrict__ A,
                            const float* __restrict__ Bm,
                            float* __restrict__ C, int nRowTiles) {
  constexpr int COLT = NCOLS / 16;
  const int w    = threadIdx.x >> 5;
  const int lane = threadIdx.x & 31;
  const int rowTilesPerBlock = (blockDim.x >> 5) / COLT;
  const int rowTile = blockIdx.x * rowTilesPerBlock + (w / COLT);
  if (rowTile >= nRowTiles) return;           // whole wave exits; EXEC stays full
  const int colTile = w % COLT;
  const int r0 = rowTile * 16, c0 = colTile * 16;
  const int m = lane & 15, half = lane >> 4;

  v8f acc = {};
  const float* ap = A  + (size_t)(r0 + m) * KDIM + 2 * half;
  const float* bp = Bm + (size_t)(2 * half) * NCOLS + c0 + m;
#pragma unroll 4
  for (int k = 0; k < KDIM; k += 4) {
    v2f a, b;
    a.x = ap[0];
    a.y = ap[1];
    b.x = bp[0];
    b.y = bp[NCOLS];
    acc = __builtin_amdgcn_wmma_f32_16x16x4_f32(
        /*neg_a=*/false, a, /*neg_b=*/false, b,
        /*c_mod=*/(short)0, acc, /*reuse_a=*/false, /*reuse_b=*/false);
    ap += 4;
    bp += 4 * NCOLS;
  }
#pragma unroll
  for (int v = 0; v < 8; ++v)
    C[(size_t)(r0 + v + 8 * half) * NCOLS + c0 + m] = acc[v];
}

// ---------------------------------------------------------------------------
// Degree (in-edges on A) via float atomics, then dinv = rsqrt(deg + 1).
// ---------------------------------------------------------------------------
__global__ void k_deg(const int* __restrict__ dst, float* __restrict__ deg, int E) {
  int e = blockIdx.x * blockDim.x + threadIdx.x;
  if (e < E) atomicAdd(&deg[dst[e]], 1.0f);
}

__global__ void k_dinv(const float* __restrict__ deg, float* __restrict__ dinv, int N) {
  int i = blockIdx.x * blockDim.x + threadIdx.x;
  if (i < N) dinv[i] = rsqrtf(deg[i] + 1.0f);
}

// ---------------------------------------------------------------------------
// Edge scatter: agg[dst] += h[src] * dinv[src]*dinv[dst].
// D/4 float4 chunks per edge, 4 f32 atomics per chunk.
// ---------------------------------------------------------------------------
template <int D>
__global__ void k_scatter(const int* __restrict__ src, const int* __restrict__ dst,
                          const float* __restrict__ dinv,
                          const float* __restrict__ h, float* __restrict__ agg,
                          long long total) {
  constexpr int C = D / 4;
  long long idx = (long long)blockIdx.x * blockDim.x + threadIdx.x;
  if (idx >= total) return;
  int e = (int)(idx / C);
  int c = (int)(idx % C);
  int s = src[e], d = dst[e];
  float coef = dinv[s] * dinv[d];
  const float4 v = *(const float4*)(h + (size_t)s * D + c * 4);
  float* out = agg + (size_t)d * D + c * 4;
  atomicAdd(out + 0, v.x * coef);
  atomicAdd(out + 1, v.y * coef);
  atomicAdd(out + 2, v.z * coef);
  atomicAdd(out + 3, v.w * coef);
}

// out = agg + h * dinv^2 + bias      (GCN self-loop term + bias)
template <int D>
__global__ void k_finish(const float* __restrict__ agg, const float* __restrict__ h,
                         const float* __restrict__ dinv, const float* __restrict__ bias,
                         float* __restrict__ out, long long total) {
  long long idx = (long long)blockIdx.x * blockDim.x + threadIdx.x;
  if (idx >= total) return;
  int n = (int)(idx / D);
  int j = (int)(idx % D);
  float di = dinv[n];
  out[idx] = agg[idx] + h[idx] * di * di + bias[j];
}

// Layer-2 finish fused with row softmax over K=32.
__global__ void k_logits_softmax(const float* __restrict__ agg2,
                                 const float* __restrict__ h2,
                                 const float* __restrict__ dinv,
                                 const float* __restrict__ b2,
                                 float* __restrict__ S, int N) {
  int n = blockIdx.x * blockDim.x + threadIdx.x;
  if (n >= N) return;
  float di = dinv[n];
  float di2 = di * di;
  float vals[32];
  float mx = -3.4e38f;
#pragma unroll
  for (int k = 0; k < 32; ++k) {
    float v = agg2[(size_t)n * 32 + k] + h2[(size_t)n * 32 + k] * di2 + b2[k];
    vals[k] = v;
    mx = fmaxf(mx, v);
  }
  float ssum = 0.0f;
#pragma unroll
  for (int k = 0; k < 32; ++k) {
    vals[k] = expf(vals[k] - mx);
    ssum += vals[k];
  }
  float inv = 1.0f / ssum;
#pragma unroll
  for (int k = 0; k < 32; ++k) S[(size_t)n * 32 + k] = vals[k] * inv;
}

// ---------------------------------------------------------------------------
// Pooling: pooled[b,k,j] = sum_{n: batch[n]==b} S[n,k] * x[n,j].
// batch is sorted -> block b binary-searches its node range, accumulates its
// 32x64 tile in registers (8 cells/thread), staging rows through LDS.
// Each output cell written exactly once (no atomics, no output init needed).
// ---------------------------------------------------------------------------
__global__ void k_pool(const float* __restrict__ S, const float* __restrict__ x,
                       const int* __restrict__ batch, float* __restrict__ pooled,
                       int N) {
  __shared__ float sS[32];
  __shared__ float sX[64];
  const int b = blockIdx.x;
  // lower_bound(batch, b)
  int lo = 0, hi = N;
  while (lo < hi) { int mid = (lo + hi) >> 1; if (batch[mid] < b) lo = mid + 1; else hi = mid; }
  const int start = lo;
  hi = N;
  while (lo < hi) { int mid = (lo + hi) >> 1; if (batch[mid] < b + 1) lo = mid + 1; else hi = mid; }
  const int end = lo;

  const int t = threadIdx.x;  // 256 threads
  float acc[8];
#pragma unroll
  for (int i = 0; i < 8; ++i) acc[i] = 0.0f;

  for (int n = start; n < end; ++n) {
    if (t < 32)      sS[t]      = S[(size_t)n * 32 + t];
    else if (t < 96) sX[t - 32] = x[(size_t)n * 64 + (t - 32)];
    __syncthreads();
#pragma unroll
    for (int i = 0; i < 8; ++i) {
      int c = t + i * 256;                 // c = k*64 + j
      acc[i] += sS[c >> 6] * sX[c & 63];
    }
    __syncthreads();
  }
#pragma unroll
  for (int i = 0; i < 8; ++i) {
    int c = t + i * 256;
    pooled[(size_t)b * 2048 + c] = acc[i];
  }
}

// ---------------------------------------------------------------------------
// Launch
// ---------------------------------------------------------------------------
extern "C" void kernel_launch(void* const* d_in, const int* in_sizes, int n_in,
                              void* d_out, int out_size, void* d_ws, size_t ws_size,
                              hipStream_t stream) {
  const float* x_in = (const float*)d_in[0];   // [N,128]
  const int*   eidx = (const int*)d_in[1];     // [2,E]
  const int*   batch = (const int*)d_in[2];    // [N]
  const float* W1 = (const float*)d_in[3];     // [128,64]
  const float* b1 = (const float*)d_in[4];     // [64]
  const float* W2 = (const float*)d_in[5];     // [64,32]
  const float* b2 = (const float*)d_in[6];     // [32]

  const int N = in_sizes[0] / 128;
  const int E = in_sizes[1] / 2;
  const int* src = eidx;
  const int* dst = eidx + E;

  // Workspace carve-up (floats)
  float* deg  = (float*)d_ws;            // N
  float* dinv = deg  + N;                // N
  float* h1   = dinv + N;                // N*64
  float* xg   = h1   + (size_t)N * 64;   // N*64
  float* agg  = xg   + (size_t)N * 64;   // N*64
  float* h2   = agg  + (size_t)N * 64;   // N*32
  float* agg2 = h2   + (size_t)N * 32;   // N*32
  float* S    = agg2 + (size_t)N * 32;   // N*32
  (void)ws_size; (void)n_in; (void)out_size;

  hipMemsetAsync(deg,  0, (size_t)N * sizeof(float), stream);
  hipMemsetAsync(agg,  0, (size_t)N * 64 * sizeof(float), stream);
  hipMemsetAsync(agg2, 0, (size_t)N * 32 * sizeof(float), stream);

  // Degree + dinv
  k_deg<<<(E + 255) / 256, 256, 0, stream>>>(dst, deg, E);
  k_dinv<<<(N + 255) / 256, 256, 0, stream>>>(deg, dinv, N);

  const int nRowTiles = N / 16;  // N = 100000 -> 6250

  // GCN layer 1: h1 = x_in @ W1 (WMMA f32), scatter, self+bias -> xg
  {
    const int rowTilesPerBlock = 2;  // 8 waves, 4 col tiles
    int grid = (nRowTiles + rowTilesPerBlock - 1) / rowTilesPerBlock;
    k_gemm_wmma<128, 64><<<grid, 256, 0, stream>>>(x_in, W1, h1, nRowTiles);
  }
  {
    long long total = (long long)E * 16;
    long long grid = (total + 255) / 256;
    k_scatter<64><<<(unsigned)grid, 256, 0, stream>>>(src, dst, dinv, h1, agg, total);
  }
  {
    long long total = (long long)N * 64;
    long long grid = (total + 255) / 256;
    k_finish<64><<<(unsigned)grid, 256, 0, stream>>>(agg, h1, dinv, b1, xg, total);
  }

  // GCN layer 2: h2 = xg @ W2 (WMMA f32), scatter, fused finish+softmax -> S
  {
    const int rowTilesPerBlock = 4;  // 8 waves, 2 col tiles
    int grid = (nRowTiles + rowTilesPerBlock - 1) / rowTilesPerBlock;
    k_gemm_wmma<64, 32><<<grid, 256, 0, stream>>>(xg, W2, h2, nRowTiles);
  }
  {
    long long total = (long long)E * 8;
    long long grid = (total + 255) / 256;
    k_scatter<32><<<(unsigned)grid, 256, 0, stream>>>(src, dst, dinv, h2, agg2, total);
  }
  k_logits_softmax<<<(N + 255) / 256, 256, 0, stream>>>(agg2, h2, dinv, b2, S, N);

  // Batch-segmented weighted pooling -> d_out [128, 32, 64]
  k_pool<<<128, 256, 0, stream>>>(S, xg, batch, (float*)d_out, N);
}